// OctonionFanoCoherence_43224550867272
// MI455X (gfx1250) — compile-verified
//
#include <hip/hip_runtime.h>
#include <cstdint>

#define THREADS 256
#define F4_PER_ELEM 14         // 56 floats = 14 float4 per batch element
#define PAD_STRIDE 60          // floats per thread in LDS: 56 + 4 pad (16B-aligned, reduced bank conflicts)

// Quaternion product (w,x,y,z)
__device__ __forceinline__ void qmul(float w1, float x1, float y1, float z1,
                                     float w2, float x2, float y2, float z2,
                                     float& rw, float& rx, float& ry, float& rz) {
    rw = w1 * w2 - x1 * x2 - y1 * y2 - z1 * z2;
    rx = w1 * x2 + x1 * w2 + y1 * z2 - z1 * y2;
    ry = w1 * y2 - x1 * z2 + y1 * w2 + z1 * x2;
    rz = w1 * z2 + x1 * y2 - y1 * x2 + z1 * w2;
}

// Octonion product via Cayley-Dickson: c = a * b
__device__ __forceinline__ void omul(const float* a, const float* b, float* c) {
    float p0, p1, p2, p3, q0, q1, q2, q3;
    // c1 = qmul(a1, b1) - qmul(conj(b2), a2)
    qmul(a[0], a[1], a[2], a[3], b[0], b[1], b[2], b[3], p0, p1, p2, p3);
    qmul(b[4], -b[5], -b[6], -b[7], a[4], a[5], a[6], a[7], q0, q1, q2, q3);
    c[0] = p0 - q0; c[1] = p1 - q1; c[2] = p2 - q2; c[3] = p3 - q3;
    // c2 = qmul(b2, a1) + qmul(a2, conj(b1))
    qmul(b[4], b[5], b[6], b[7], a[0], a[1], a[2], a[3], p0, p1, p2, p3);
    qmul(a[4], a[5], a[6], a[7], b[0], -b[1], -b[2], -b[3], q0, q1, q2, q3);
    c[4] = p0 + q0; c[5] = p1 + q1; c[6] = p2 + q2; c[7] = p3 + q3;
}

__global__ __launch_bounds__(THREADS)
void OctonionFanoCoherence_kernel(const float* __restrict__ in,
                                  const float* __restrict__ logsens,
                                  float* __restrict__ out, int B) {
    __shared__ __align__(16) float smem[THREADS * PAD_STRIDE];   // 61,440 B

    const int tid  = threadIdx.x;
    const unsigned lane = tid & 31;
    const unsigned wv   = tid >> 5;
    const unsigned blockElem = (unsigned)blockIdx.x * THREADS;

    // ---- Stage 1: coalesced async copy (GVS mode: SGPR base + 32-bit VGPR byte offset) ----
    // Wave wv owns elements [blockElem + wv*32, +32) = 448 contiguous float4s.
    // Total input is B*224 bytes (< 2^31 for this problem), so 32-bit offsets suffice.
    const unsigned total4    = (unsigned)B * F4_PER_ELEM;
    const unsigned waveBase4 = (blockElem + wv * 32u) * F4_PER_ELEM;
    const uint64_t sbase     = (uint64_t)(const void*)in;

    #pragma unroll
    for (int i = 0; i < F4_PER_ELEM; ++i) {
        unsigned gw = (unsigned)i * 32u + lane;        // float4 index within wave chunk [0,448)
        unsigned g4 = waveBase4 + gw;
        g4 = (g4 < total4) ? g4 : (total4 - 1u);       // branchless tail clamp (v_min)
        unsigned goff = g4 << 4;                       // byte offset into input
        unsigned tw = gw / F4_PER_ELEM;                // thread-in-wave consuming this float4
        unsigned f4 = gw - tw * F4_PER_ELEM;           // which of its 14 float4s
        unsigned ldsOff = (unsigned)(size_t)(&smem[(wv * 32u + tw) * PAD_STRIDE + f4 * 4u]);
        asm volatile("global_load_async_to_lds_b128 %0, %1, %2 th:TH_LOAD_NT"
                     :: "v"(ldsOff), "v"(goff), "s"(sbase) : "memory");
    }
    asm volatile("s_wait_asynccnt 0" ::: "memory");
    // Each wave reads only LDS it wrote itself -> no s_barrier needed.

    // ---- Stage 2: gather 7x8 state, normalize rows ----
    float st[7][8];
    const float4* p = (const float4*)&smem[tid * PAD_STRIDE];
    #pragma unroll
    for (int r = 0; r < 7; ++r) {
        float4 lo = p[r * 2 + 0];
        float4 hi = p[r * 2 + 1];
        st[r][0] = lo.x; st[r][1] = lo.y; st[r][2] = lo.z; st[r][3] = lo.w;
        st[r][4] = hi.x; st[r][5] = hi.y; st[r][6] = hi.z; st[r][7] = hi.w;
    }
    #pragma unroll
    for (int r = 0; r < 7; ++r) {
        float n2 = 0.0f;
        #pragma unroll
        for (int c = 0; c < 8; ++c) n2 = fmaf(st[r][c], st[r][c], n2);
        // 1/max(sqrt(n2),1e-12) == rsqrt(max(n2,1e-24)) -> single v_rsq_f32
        float inv = __builtin_amdgcn_rsqf(fmaxf(n2, 1e-24f));
        #pragma unroll
        for (int c = 0; c < 8; ++c) st[r][c] *= inv;
    }

    // ---- Stage 3: Fano-line octonion products + squared error ----
    // FANO row l = (l, (l+1)%7, (l+3)%7)
    float err = 0.0f;
    #pragma unroll
    for (int l = 0; l < 7; ++l) {
        const float* xi = st[l];
        const float* xj = st[(l + 1) % 7];
        const float* xk = st[(l + 3) % 7];
        float c[8];
        omul(xi, xj, c);
        #pragma unroll
        for (int m = 0; m < 8; ++m) {
            float d = c[m] - xk[m];
            err = fmaf(d, d, err);
        }
    }

    // ---- Stage 4: phi ----
    float es  = __expf(logsens[0]);
    float phi = -__logf(err * (1.0f / 7.0f) + 1e-8f) * es;
    phi = fminf(fmaxf(phi, 0.0f), 10.0f);

    unsigned elem = blockElem + (unsigned)tid;
    if (elem < (unsigned)B) out[elem] = phi;
}

extern "C" void kernel_launch(void* const* d_in, const int* in_sizes, int n_in,
                              void* d_out, int out_size, void* d_ws, size_t ws_size,
                              hipStream_t stream) {
    const float* states  = (const float*)d_in[0];   // (B,7,8) fp32
    const float* logsens = (const float*)d_in[1];   // scalar fp32
    float*       out     = (float*)d_out;           // (B,) fp32
    (void)d_ws; (void)ws_size; (void)n_in; (void)out_size;

    int B = in_sizes[0] / 56;
    int blocks = (B + THREADS - 1) / THREADS;
    hipLaunchKernelGGL(OctonionFanoCoherence_kernel,
                       dim3(blocks), dim3(THREADS), 0, stream,
                       states, logsens, out, B);
}